// GNNEncoder_24945170055874
// MI455X (gfx1250) — compile-verified
//
#include <hip/hip_runtime.h>
#include <hip/hip_bf16.h>
#include <stdint.h>

// ---------------------------------------------------------------------------
// GCN 2-layer encoder for MI455X (gfx1250, wave32).
// - GEMMs: V_WMMA_F32_16X16X4_F32 (fp32 in/out; GEMM is compute-trivial vs
//   the bandwidth-bound edge aggregation, so full fp32 precision is free).
// - X tiles staged to LDS via GLOBAL_LOAD_ASYNC_TO_LDS_B128 (ASYNCcnt path).
// - Each wave computes two 16x16 row-tiles sharing B fragments (halves W
//   traffic, 2 WMMA per B load).
// ---------------------------------------------------------------------------

typedef __attribute__((ext_vector_type(2))) float v2f;
typedef __attribute__((ext_vector_type(8))) float v8f;

#define DHID 128
#define NEG_SLOPE 0.01f
#define DROP_P 0.1f

// ---------------- degree / norm -------------------------------------------

__global__ void init_deg_kernel(float* deg, int n) {
    int i = blockIdx.x * blockDim.x + threadIdx.x;
    if (i < n) deg[i] = 1.0f;   // self-loop
}

__global__ void count_deg_kernel(const long long* __restrict__ dst, float* deg, int e) {
    int i = blockIdx.x * blockDim.x + threadIdx.x;
    if (i < e) atomicAdd(&deg[(int)dst[i]], 1.0f);
}

__global__ void rsqrt_kernel(float* deg, int n) {
    int i = blockIdx.x * blockDim.x + threadIdx.x;
    if (i < n) deg[i] = rsqrtf(deg[i]);   // in place: deg -> deg_inv_sqrt
}

// ---------------- fp32 WMMA GEMM: Y[N,128] = X[N,128] @ W[128,128] ---------
// Block = 256 threads = 8 waves, covers 32 rows x 128 cols.
// Wave w -> cols [16w, 16w+16), two row tiles (rows 0-15 and 16-31 of block).
// K-loop: 32 steps of v_wmma_f32_16x16x4_f32 per tile, B frags shared.

__global__ __launch_bounds__(256) void gemm_wmma_f32_kernel(
    const float* __restrict__ X, const float* __restrict__ W,
    float* __restrict__ Y, int nrows) {

    __shared__ float xt[32 * DHID];   // 16 KB tile of X

    const int rowBase = blockIdx.x * 32;
    const int lane = threadIdx.x & 31;
    const int wave = threadIdx.x >> 5;

    // --- async stage 32x128 X tile into LDS (4 x b128 per thread) ---------
    {
        unsigned ldsaddr = (unsigned)(uintptr_t)(&xt[0]) + threadIdx.x * 16u;
        unsigned long long gaddr =
            (unsigned long long)(X + (size_t)rowBase * DHID + threadIdx.x * 4u);
        asm volatile(
            "global_load_async_to_lds_b128 %0, %1, off\n\t"
            "global_load_async_to_lds_b128 %0, %1, off offset:4096\n\t"
            "global_load_async_to_lds_b128 %0, %1, off offset:8192\n\t"
            "global_load_async_to_lds_b128 %0, %1, off offset:12288"
            :: "v"(ldsaddr), "v"(gaddr) : "memory");
        asm volatile("s_wait_asynccnt 0x0" ::: "memory");
    }
    __syncthreads();

    const int colBase = wave * 16;
    const int m = lane & 15;          // A row within tile / B-C col within tile
    const int koff = (lane >> 4) * 2; // K sub-offset per half-wave

    v8f acc0 = {};
    v8f acc1 = {};
    #pragma unroll 4
    for (int kk = 0; kk < DHID; kk += 4) {
        // A fragments from LDS (8B-aligned pairs), two row tiles
        const float* ap0 = xt + m * DHID + kk + koff;
        const float* ap1 = ap0 + 16 * DHID;
        v2f a0; a0.x = ap0[0]; a0.y = ap0[1];
        v2f a1; a1.x = ap1[0]; a1.y = ap1[1];
        // B fragment from global (L2-resident W), shared by both tiles
        const float* bp = W + (size_t)(kk + koff) * DHID + colBase + m;
        v2f b; b.x = bp[0]; b.y = bp[DHID];
        acc0 = __builtin_amdgcn_wmma_f32_16x16x4_f32(
            false, a0, false, b, (short)0, acc0, false, false);
        acc1 = __builtin_amdgcn_wmma_f32_16x16x4_f32(
            false, a1, false, b, (short)0, acc1, false, false);
    }

    // scatter C: VGPR r -> row (r + 8*(lane>>4)), col (lane&15)
    const int outRow = rowBase + (lane >> 4) * 8;
    const int outCol = colBase + m;
    #pragma unroll
    for (int r = 0; r < 8; ++r) {
        Y[(size_t)(outRow + r) * DHID + outCol] = acc0[r];
        Y[(size_t)(outRow + 16 + r) * DHID + outCol] = acc1[r];
    }
}

// ---------------- aggregation ---------------------------------------------

__global__ void zero_kernel(float4* p, int n4) {
    int i = blockIdx.x * blockDim.x + threadIdx.x;
    if (i < n4) p[i] = make_float4(0.f, 0.f, 0.f, 0.f);
}

// one thread = (edge, 4 channels); 32 threads per edge
__global__ __launch_bounds__(256) void scatter_kernel(
    const long long* __restrict__ src, const long long* __restrict__ dst,
    const float* __restrict__ h, const float* __restrict__ dinv,
    float* __restrict__ agg, int e) {

    int t = blockIdx.x * blockDim.x + threadIdx.x;
    int edge = t >> 5;
    if (edge >= e) return;
    int c = (t & 31) << 2;
    int s = (int)src[edge];
    int d = (int)dst[edge];
    float norm = dinv[s] * dinv[d];
    float4 hv = *(const float4*)(h + (size_t)s * DHID + c);
    float* base = agg + (size_t)d * DHID + c;
    atomicAdd(base + 0, hv.x * norm);
    atomicAdd(base + 1, hv.y * norm);
    atomicAdd(base + 2, hv.z * norm);
    atomicAdd(base + 3, hv.w * norm);
}

__device__ __forceinline__ unsigned mix_hash(unsigned a, unsigned b) {
    unsigned h = a * 0x9E3779B9u ^ (b + 0x7F4A7C15u + (a << 6) + (a >> 2));
    h ^= h >> 16; h *= 0x85EBCA6Bu;
    h ^= h >> 13; h *= 0xC2B2AE35u;
    h ^= h >> 16;
    return h;
}

// out = agg + h*dinv^2 + bias, optional leaky-relu + deterministic dropout.
// one thread = (node, 4 channels), writes float4.
__global__ __launch_bounds__(256) void combine_kernel(
    const float* __restrict__ agg, const float* __restrict__ h,
    const float* __restrict__ dinv, const float* __restrict__ bias,
    float* __restrict__ out, int n, int apply_act) {

    int t = blockIdx.x * blockDim.x + threadIdx.x;
    int node = t >> 5;
    if (node >= n) return;
    int c = (t & 31) << 2;
    float selfn = dinv[node];
    selfn *= selfn;   // 1/deg

    float4 av = *(const float4*)(agg + (size_t)node * DHID + c);
    float4 hv = *(const float4*)(h + (size_t)node * DHID + c);
    float4 bv = *(const float4*)(bias + c);

    float v[4];
    v[0] = av.x + hv.x * selfn + bv.x;
    v[1] = av.y + hv.y * selfn + bv.y;
    v[2] = av.z + hv.z * selfn + bv.z;
    v[3] = av.w + hv.w * selfn + bv.w;

    if (apply_act) {
        const unsigned thresh = (unsigned)(DROP_P * 16777216.0f);  // of 2^24
        const float scale = 1.0f / (1.0f - DROP_P);
        #pragma unroll
        for (int j = 0; j < 4; ++j) {
            float x = v[j];
            x = (x > 0.f) ? x : NEG_SLOPE * x;                      // leaky_relu
            unsigned r = mix_hash((unsigned)node, (unsigned)(c + j)) & 0xFFFFFFu;
            v[j] = (r >= thresh) ? x * scale : 0.0f;                // dropout
        }
    }
    *(float4*)(out + (size_t)node * DHID + c) = make_float4(v[0], v[1], v[2], v[3]);
}

// ---------------------------------------------------------------------------

extern "C" void kernel_launch(void* const* d_in, const int* in_sizes, int n_in,
                              void* d_out, int out_size, void* d_ws, size_t ws_size,
                              hipStream_t stream) {
    const float*      x  = (const float*)d_in[0];
    const long long*  ei = (const long long*)d_in[1];   // int64 edge_index [2,E]
    const float*      W1 = (const float*)d_in[2];
    const float*      b1 = (const float*)d_in[3];
    const float*      W2 = (const float*)d_in[4];
    const float*      b2 = (const float*)d_in[5];
    float* out = (float*)d_out;

    const int N = in_sizes[0] / DHID;       // 100000
    const int E = in_sizes[1] / 2;          // 1600000
    const long long* src = ei;
    const long long* dst = ei + E;

    // workspace layout
    char* ws = (char*)d_ws;
    float* dinv = (float*)ws;                                  // N floats
    float* hbuf = (float*)(ws + (1ull << 20));                 // N*128 floats
    float* abuf = (float*)(ws + (1ull << 20) + (50ull << 20)); // N*128 floats

    const int B = 256;
    const int nh4 = N * (DHID / 4);               // float4 count of N x H
    const int edgeThreads = E * 32;               // (edge, 4ch) threads
    const int nodeThreads = N * 32;

    // --- degree / norm ---
    init_deg_kernel<<<(N + B - 1) / B, B, 0, stream>>>(dinv, N);
    count_deg_kernel<<<(E + B - 1) / B, B, 0, stream>>>(dst, dinv, E);
    rsqrt_kernel<<<(N + B - 1) / B, B, 0, stream>>>(dinv, N);

    // --- layer 1 ---
    gemm_wmma_f32_kernel<<<N / 32, B, 0, stream>>>(x, W1, hbuf, N);
    zero_kernel<<<(nh4 + B - 1) / B, B, 0, stream>>>((float4*)abuf, nh4);
    scatter_kernel<<<(edgeThreads + B - 1) / B, B, 0, stream>>>(src, dst, hbuf, dinv, abuf, E);
    combine_kernel<<<(nodeThreads + B - 1) / B, B, 0, stream>>>(
        abuf, hbuf, dinv, b1, abuf, N, /*apply_act=*/1);   // in-place -> activations

    // --- layer 2 ---
    gemm_wmma_f32_kernel<<<N / 32, B, 0, stream>>>(abuf, W2, hbuf, N);
    zero_kernel<<<(nh4 + B - 1) / B, B, 0, stream>>>((float4*)abuf, nh4);
    scatter_kernel<<<(edgeThreads + B - 1) / B, B, 0, stream>>>(src, dst, hbuf, dinv, abuf, E);
    combine_kernel<<<(nodeThreads + B - 1) / B, B, 0, stream>>>(
        abuf, hbuf, dinv, b2, out, N, /*apply_act=*/0);
}